// MultiTaskWithCRF_41618233099118
// MI455X (gfx1250) — compile-verified
//
#include <hip/hip_runtime.h>
#include <hip/hip_bf16.h>
#include <stdint.h>

// ---------------------------------------------------------------------------
// CRF NLL on MI455X (gfx1250, wave32).
//  kernel 1: per-(batch, half-sequence) chunked forward DP in exp domain,
//            emissions staged via global_load_async_to_lds_b64 (ASYNCcnt),
//            LDS layout padded to 194 dwords/chunk (bank-conflict free).
//  kernel 2: join the two half-sequence 3x3 log-semiring matrices per batch,
//            add end transitions, produce per-batch (log_z - gold_score).
//  kernel 3: deterministic 1024-way reduction; the 32 lane partials are
//            summed on the matrix pipe via v_wmma_f32_16x16x4_f32 with an
//            all-ones B matrix (row-sum trick).
// ---------------------------------------------------------------------------

typedef __attribute__((ext_vector_type(2))) float v2f;
typedef __attribute__((ext_vector_type(8))) float v8f;

#define B_DIM 1024
#define S_DIM 8192
#define HALF_S 4096
#define CHUNKS 64      // chunks per half-sequence
#define CLEN 64        // steps per chunk
#define LDS_STRIDE 194 // dwords per chunk in LDS: 192 data + 2 pad
                       // 194 % 64 == 2  -> 32 lanes hit 32 distinct banks
                       // 194*4 = 776 B  -> 8-byte aligned chunk bases (b64 DMA ok)

__device__ __forceinline__ unsigned lds_lo32(const void* p) {
  // low 32 bits of a generic pointer to LDS == LDS byte address
  return (unsigned)(uintptr_t)p;
}

__device__ __forceinline__ void async_ld_b64(unsigned lds_byte, const void* gptr) {
  unsigned long long ga = (unsigned long long)(uintptr_t)gptr;
  asm volatile("global_load_async_to_lds_b64 %0, %1, off"
               :: "v"(lds_byte), "v"(ga)
               : "memory");
}

__device__ __forceinline__ void wait_async0() {
  asm volatile("s_wait_asynccnt 0x0" ::: "memory");
}

__device__ __forceinline__ float lse3(float a, float b, float c) {
  float m = fmaxf(a, fmaxf(b, c));
  return m + __logf(__expf(a - m) + __expf(b - m) + __expf(c - m));
}

// ---------------------------------------------------------------------------
// Kernel 1: grid = 2048 blocks (batch * 2 halves), block = 64 threads.
// Each thread owns one 64-step chunk; outputs a 3x3 log-transfer matrix +
// gold-score partial; block tree-combines its 64 chunks in LDS.
// ---------------------------------------------------------------------------
__global__ __launch_bounds__(64) void crf_chunk_kernel(
    const float* __restrict__ em, const int* __restrict__ lab,
    const float* __restrict__ startT, const float* __restrict__ trans,
    const float* __restrict__ endT, float* __restrict__ out10) {
  __shared__ float sE[CHUNKS * LDS_STRIDE]; // 49,664 B emissions (padded)
  __shared__ float sC[16];                  // T[9], start[3], end[3]
  __shared__ float sM[CHUNKS][10];          // tree scratch: 9 matrix + gold

  const int tid = threadIdx.x;              // chunk id 0..63
  const int blk = blockIdx.x;               // 0..2047
  const int b = blk >> 1;
  const int h = blk & 1;

  if (tid < 9)       sC[tid] = trans[tid];
  else if (tid < 12) sC[tid] = startT[tid - 9];
  else if (tid < 15) sC[tid] = endT[tid - 12];

  // ---- phase A: async DMA emissions (half sequence) into padded LDS ----
  const float* gsrc = em + ((size_t)b * (S_DIM * 3) + (size_t)h * (HALF_S * 3));
  const unsigned sE0 = lds_lo32(&sE[0]);
#pragma unroll 4
  for (int i = 0; i < 96; ++i) {
    int q  = i * 64 + tid;       // b64 unit index, contiguous across lanes
    int ch = q / 96;             // destination chunk
    int r  = q - ch * 96;        // b64 offset within chunk
    async_ld_b64(sE0 + (unsigned)(ch * LDS_STRIDE + r * 2) * 4u,
                 gsrc + (size_t)q * 2);
  }

  const int t0 = h * HALF_S + tid * CLEN;               // first step of chunk
  const int* lb = lab + (size_t)b * S_DIM + t0;
  __builtin_prefetch(lb, 0, 1);                         // global_prefetch label stream

  wait_async0();
  __syncthreads();

  // ---- phase B: exp-domain chunk DP (3 basis rows) + gold partials ----
  float eT[9];
#pragma unroll
  for (int i = 0; i < 9; ++i) eT[i] = __expf(sC[i]);

  float v[3][3], ls[3];
#pragma unroll
  for (int r = 0; r < 3; ++r) {
    ls[r] = 0.f;
#pragma unroll
    for (int j = 0; j < 3; ++j) v[r][j] = (r == j) ? 1.f : 0.f;
  }

  const unsigned base = (unsigned)tid * LDS_STRIDE;
  float gold = 0.f;
  int labprev = 0;
  int kstart = 0;

  if (t0 == 0) {
    // very first chunk: row 0 carries alpha0 = start + e0 (no transition in)
    float s0 = sC[9], s1 = sC[10], s2 = sC[11];
    float m = fmaxf(s0, fmaxf(s1, s2));
    v[0][0] = __expf(s0 - m); v[0][1] = __expf(s1 - m); v[0][2] = __expf(s2 - m);
    ls[0] = m;
    float e0 = sE[base + 0], e1 = sE[base + 1], e2 = sE[base + 2];
    v[0][0] *= __expf(e0); v[0][1] *= __expf(e1); v[0][2] *= __expf(e2);
    int l0 = lb[0];
    gold += (l0 == 0 ? e0 : (l0 == 1 ? e1 : e2)) + sC[9 + l0]; // emission + start
    labprev = l0;
    kstart = 1;
  } else {
    labprev = lab[(size_t)b * S_DIM + t0 - 1];
  }

#pragma unroll 2
  for (int k = kstart; k < CLEN; ++k) {
    float e0 = sE[base + k * 3 + 0];
    float e1 = sE[base + k * 3 + 1];
    float e2 = sE[base + k * 3 + 2];
    float x0 = __expf(e0), x1 = __expf(e1), x2 = __expf(e2);
#pragma unroll
    for (int r = 0; r < 3; ++r) {
      float n0 = v[r][0] * eT[0] + v[r][1] * eT[3] + v[r][2] * eT[6];
      float n1 = v[r][0] * eT[1] + v[r][1] * eT[4] + v[r][2] * eT[7];
      float n2 = v[r][0] * eT[2] + v[r][1] * eT[5] + v[r][2] * eT[8];
      v[r][0] = n0 * x0; v[r][1] = n1 * x1; v[r][2] = n2 * x2;
    }
    int lc = lb[k];
    float ecur = (lc == 0 ? e0 : (lc == 1 ? e1 : e2));
    gold += ecur + sC[labprev * 3 + lc];
    labprev = lc;
    if ((k & 7) == 7) { // periodic renormalization keeps f32 in range
#pragma unroll
      for (int r = 0; r < 3; ++r) {
        float s = v[r][0] + v[r][1] + v[r][2];
        float inv = 1.0f / s;
        v[r][0] *= inv; v[r][1] *= inv; v[r][2] *= inv;
        ls[r] += __logf(s);
      }
    }
  }
  if (t0 + CLEN == S_DIM) gold += sC[12 + labprev]; // end transition (gold path)

  // back to log domain
#pragma unroll
  for (int r = 0; r < 3; ++r)
#pragma unroll
    for (int j = 0; j < 3; ++j)
      sM[tid][r * 3 + j] = ls[r] + __logf(fmaxf(v[r][j], 1e-37f));
  sM[tid][9] = gold;

  // ---- tree combine (log-semiring matmul) over the 64 chunks ----
  for (int st = 1; st < CHUNKS; st <<= 1) {
    __syncthreads();
    if ((tid & (2 * st - 1)) == 0) {
      float A[10], Bm[10];
#pragma unroll
      for (int i = 0; i < 10; ++i) { A[i] = sM[tid][i]; Bm[i] = sM[tid + st][i]; }
      float Cm[9];
#pragma unroll
      for (int i = 0; i < 3; ++i)
#pragma unroll
        for (int j = 0; j < 3; ++j)
          Cm[i * 3 + j] = lse3(A[i * 3 + 0] + Bm[0 + j],
                               A[i * 3 + 1] + Bm[3 + j],
                               A[i * 3 + 2] + Bm[6 + j]);
#pragma unroll
      for (int i = 0; i < 9; ++i) sM[tid][i] = Cm[i];
      sM[tid][9] = A[9] + Bm[9];
    }
  }
  __syncthreads();
  if (tid == 0) {
#pragma unroll
    for (int i = 0; i < 10; ++i) out10[(size_t)blk * 10 + i] = sM[0][i];
  }
}

// ---------------------------------------------------------------------------
// Kernel 2: join the two halves per batch, fold end transitions,
// emit per-batch (log_z - gold_score).
// ---------------------------------------------------------------------------
__global__ __launch_bounds__(256) void crf_finish_kernel(
    const float* __restrict__ ws10, const float* __restrict__ endT,
    float* __restrict__ part) {
  int bidx = blockIdx.x * 256 + threadIdx.x;
  if (bidx >= B_DIM) return;
  const float* A  = ws10 + (size_t)(2 * bidx) * 10; // first half (row0 = alpha)
  const float* Bm = A + 10;                          // second half (full matrix)
  float a[3];
#pragma unroll
  for (int j = 0; j < 3; ++j)
    a[j] = lse3(A[0] + Bm[0 + j], A[1] + Bm[3 + j], A[2] + Bm[6 + j]);
  float logz = lse3(a[0] + endT[0], a[1] + endT[1], a[2] + endT[2]);
  float gold = A[9] + Bm[9];
  part[bidx] = logz - gold; // contribution to -log_likelihood
}

// ---------------------------------------------------------------------------
// Kernel 3: deterministic reduction of 1024 per-batch values.
// 32 lane partials are summed on the matrix pipe:
// D = A x ones, A[16x4] f32 holds the partials -> row sums.
// ---------------------------------------------------------------------------
__global__ __launch_bounds__(32) void crf_reduce_kernel(
    const float* __restrict__ part, float* __restrict__ out) {
  int l = threadIdx.x;
  float s = 0.f;
#pragma unroll
  for (int i = 0; i < 32; ++i) s += part[l + 32 * i]; // coalesced, fixed order
  v2f a;  a.x = s;   a.y = 0.f;   // A[m][k]: lane<16 -> K0/K1, lane>=16 -> K2/K3
  v2f bm; bm.x = 1.f; bm.y = 1.f; // all-ones B  => D[m][n] = sum_k A[m][k]
  v8f c = {};
  v8f d = __builtin_amdgcn_wmma_f32_16x16x4_f32(
      /*neg_a=*/false, a, /*neg_b=*/false, bm,
      /*c_mod=*/(short)0, c, /*reuse_a=*/false, /*reuse_b=*/false);
  float t = d[0] + d[1] + d[2] + d[3] + d[4] + d[5] + d[6] + d[7];
  float o = __shfl(t, 16, 32); // lane16 holds rows 8..15 of column 0
  if (l == 0) out[0] = t + o;
}

// ---------------------------------------------------------------------------
extern "C" void kernel_launch(void* const* d_in, const int* in_sizes, int n_in,
                              void* d_out, int out_size, void* d_ws, size_t ws_size,
                              hipStream_t stream) {
  (void)in_sizes; (void)n_in; (void)out_size; (void)ws_size;
  const float* em  = (const float*)d_in[0];
  const int*   lab = (const int*)d_in[1];
  const float* st  = (const float*)d_in[2];
  const float* tr  = (const float*)d_in[3];
  const float* en  = (const float*)d_in[4];
  float* ws10 = (float*)d_ws;            // [2048][10] half-sequence results
  float* part = ws10 + 2048 * 10;        // [1024] per-batch partials

  crf_chunk_kernel<<<2048, 64, 0, stream>>>(em, lab, st, tr, en, ws10);
  crf_finish_kernel<<<(B_DIM + 255) / 256, 256, 0, stream>>>(ws10, en, part);
  crf_reduce_kernel<<<1, 32, 0, stream>>>(part, (float*)d_out);
}